// Model_41351945126184
// MI455X (gfx1250) — compile-verified
//
#include <hip/hip_runtime.h>
#include <hip/hip_bf16.h>

// Problem constants (match reference)
#define BB 8
#define NN 4096
#define DD 3
#define EE 64
#define KK 32

typedef __attribute__((ext_vector_type(2))) float v2f;
typedef __attribute__((ext_vector_type(8))) float v8f;

// ---------------- wave32 64-bit xor-shuffle via ds_swizzle (no addr VGPR) ----
// bitmask mode: new_lane = ((lane & and_mask) | or_mask) ^ xor_mask
// imm = xor<<10 | or<<5 | and ; and=0x1f, or=0
template <int XM>
__device__ __forceinline__ unsigned long long swzxor64(unsigned long long v) {
    constexpr int imm = (XM << 10) | 0x1f;
    unsigned int lo = (unsigned int)__builtin_amdgcn_ds_swizzle((int)(unsigned int)v, imm);
    unsigned int hi = (unsigned int)__builtin_amdgcn_ds_swizzle((int)(unsigned int)(v >> 32), imm);
    return ((unsigned long long)hi << 32) | (unsigned long long)lo;
}

// One bitonic compare-exchange stage at distance J inside sort blocks of size K.
// K==0 means "always ascending" (used for the final sort pass and merges).
template <int J, int K>
__device__ __forceinline__ unsigned long long cmpx(unsigned long long v, int lane) {
    unsigned long long o = swzxor64<J>(v);
    bool up    = (K == 0) ? true : ((lane & K) == 0);
    bool lower = ((lane & J) == 0);
    unsigned long long mn = (v < o) ? v : o;
    unsigned long long mx = (v < o) ? o : v;
    return (lower == up) ? mn : mx;
}

// Full bitonic sort of 32 keys (one per lane), ascending across lanes.
__device__ __forceinline__ unsigned long long warp_sort32_asc(unsigned long long v, int lane) {
    v = cmpx<1, 2>(v, lane);
    v = cmpx<2, 4>(v, lane);  v = cmpx<1, 4>(v, lane);
    v = cmpx<4, 8>(v, lane);  v = cmpx<2, 8>(v, lane);  v = cmpx<1, 8>(v, lane);
    v = cmpx<8, 16>(v, lane); v = cmpx<4, 16>(v, lane); v = cmpx<2, 16>(v, lane);
    v = cmpx<1, 16>(v, lane);
    v = cmpx<16, 0>(v, lane); v = cmpx<8, 0>(v, lane);  v = cmpx<4, 0>(v, lane);
    v = cmpx<2, 0>(v, lane);  v = cmpx<1, 0>(v, lane);
    return v;
}

// Half-cleaner cascade: sorts a bitonic sequence ascending (5 stages).
__device__ __forceinline__ unsigned long long warp_merge32_asc(unsigned long long v, int lane) {
    v = cmpx<16, 0>(v, lane); v = cmpx<8, 0>(v, lane); v = cmpx<4, 0>(v, lane);
    v = cmpx<2, 0>(v, lane);  v = cmpx<1, 0>(v, lane);
    return v;
}

// =====================================================================
// Kernel A: fused distance (FP32 WMMA 16x16x4) + per-row bitonic top-32
// grid = B * N/16 blocks, 512 threads (16 waves).
// Wave w computes the 16x16 tile for columns [c0+16w, c0+16w+16) each chunk,
// then owns query row w of the tile for selection.
// =====================================================================
__global__ __launch_bounds__(512) void knn_topk_kernel(const float* __restrict__ x,
                                                       int* __restrict__ knn) {
    const int bid  = blockIdx.x;         // 0 .. B*(N/16)-1
    const int b    = bid >> 8;           // / (N/16)
    const int q0   = (bid & 255) << 4;   // query tile start in [0,N)
    const float* xb = x + (size_t)b * NN * DD;

    const int tid  = threadIdx.x;
    const int wave = tid >> 5;
    const int lane = tid & 31;
    const int half = lane >> 4;          // 0: K=0,1 / M=0..7 side; 1: K=2,3 / M=8..15 side
    const int l16  = lane & 15;

    __shared__ float distLDS[16][256];   // 16 KB chunk of the distance matrix
    __shared__ float sqs[16];            // |x_query|^2

    if (tid < 16) {
        const float* q = xb + (size_t)(q0 + tid) * DD;
        sqs[tid] = fmaf(q[0], q[0], fmaf(q[1], q[1], q[2] * q[2]));
    }
    __syncthreads();

    // A operand: row m = l16, A[m] = (-2x0, -2x1, -2x2, 1) ; branch-free selects
    v2f a;
    {
        const float* q = xb + (size_t)(q0 + l16) * DD;
        const float q0v = q[0], q1v = q[1], q2v = q[2];
        a[0] = -2.0f * (half ? q2v : q0v);
        a[1] = half ? 1.0f : (-2.0f * q1v);
    }
    // C init: c[v] = |x_{M}|^2, M = v + 8*half
    v8f cinit;
#pragma unroll
    for (int v = 0; v < 8; ++v) cinit[v] = sqs[v + 8 * half];

    // selection state: sorted ascending across lanes (lane 0 = nearest)
    unsigned long long best = ((unsigned long long)0x7F7FFFFFull << 32); // FLT_MAX | idx 0
    unsigned int kmaxbits = 0x7F7FFFFFu;

    for (int c0 = 0; c0 < NN; c0 += 256) {
        // ---- distance tile via WMMA (all waves, EXEC all-ones) ----
        const int col = c0 + (wave << 4) + l16;
        const float* yp = xb + (size_t)col * DD;
        const float y0 = yp[0], y1 = yp[1], y2 = yp[2];
        const float sq = fmaf(y0, y0, fmaf(y1, y1, y2 * y2));
        v2f bb;
        bb[0] = half ? y2 : y0;
        bb[1] = half ? sq : y1;

        // prefetch next chunk's candidate coordinates
        {
            int pcol = col + 256;
            pcol = (pcol < NN) ? pcol : col;
            __builtin_prefetch(xb + (size_t)pcol * DD, 0, 0);
        }

        v8f d = __builtin_amdgcn_wmma_f32_16x16x4_f32(
            false, a, false, bb, (short)0, cinit, false, false);

#pragma unroll
        for (int v = 0; v < 8; ++v) {
            // clamp tiny negative self-distances so float bits order as uint
            distLDS[v + 8 * half][(wave << 4) + l16] = fmaxf(d[v], 0.0f);
        }
        __syncthreads();

        // ---- per-row top-32 merge: wave owns query row 'wave' ----
#pragma unroll
        for (int bch = 0; bch < 8; ++bch) {
            const int ci = (bch << 5) + lane;
            const unsigned int dbits = __float_as_uint(distLDS[wave][ci]);
            if (__ballot(dbits < kmaxbits) != 0ull) {
                unsigned long long cand =
                    ((unsigned long long)dbits << 32) | (unsigned int)(c0 + ci);
                cand = warp_sort32_asc(cand, lane);    // ascending
                cand = swzxor64<31>(cand);             // -> descending
                unsigned long long lo = (best < cand) ? best : cand; // 32 smallest (bitonic)
                best = warp_merge32_asc(lo, lane);     // restore ascending
                kmaxbits = (unsigned int)__builtin_amdgcn_readlane(
                    (int)(unsigned int)(best >> 32), 31);   // uniform threshold (SGPR)
            }
        }
        __syncthreads();
    }

    // lane l writes the l-th nearest neighbor index
    knn[((size_t)b * NN + q0 + wave) * KK + lane] = (int)(best & 0xFFFFFFFFull);
}

// =====================================================================
// Kernel B0: T = x @ theta_w  via FP32 WMMA 16x16x4 (K slot 3 zero-padded)
// 256 threads = 8 waves; wave pair (2w,2w+1) shares a 16-point tile, each
// wave does 2 of the 4 E-tiles. grid = B*N/64 blocks.
// =====================================================================
__global__ __launch_bounds__(256) void proj_kernel(const float* __restrict__ x,
                                                   const float* __restrict__ tw,
                                                   float* __restrict__ Tm) {
    const int tid  = threadIdx.x;
    const int wave = tid >> 5;           // 0..7
    const int lane = tid & 31;
    const int half = lane >> 4;
    const int l16  = lane & 15;
    const int p0   = (blockIdx.x << 6) + ((wave >> 1) << 4);  // 16-point tile
    const int epair = wave & 1;                               // which 2 E-tiles

    v2f a;
    {
        const float* xp = x + (size_t)(p0 + l16) * DD;
        const float x0 = xp[0], x1 = xp[1], x2 = xp[2];
        a[0] = half ? x2 : x0;
        a[1] = half ? 0.0f : x1;
    }
    v8f c = {};

#pragma unroll
    for (int t = 0; t < 2; ++t) {
        const int e = ((2 * epair + t) << 4) + l16;
        v2f bb;
        bb[0] = half ? tw[2 * EE + e] : tw[e];
        bb[1] = half ? 0.0f           : tw[EE + e];

        v8f d = __builtin_amdgcn_wmma_f32_16x16x4_f32(
            false, a, false, bb, (short)0, c, false, false);

#pragma unroll
        for (int v = 0; v < 8; ++v)
            Tm[(size_t)(p0 + v + 8 * half) * EE + e] = d[v];
    }
}

// =====================================================================
// Kernel B1: out[p,e] = max_k T[nbr_k, e] + x_p @ (phi_w - theta_w)[:,e]
//                       + phi_b[e] + theta_b[e]
// 256 threads = 4 points per block (64 threads each, thread = channel e).
// T reads are fully coalesced 256B segments per neighbor row.
// =====================================================================
__global__ __launch_bounds__(256) void gather_max_kernel(const int* __restrict__ knn,
                                                         const float* __restrict__ Tm,
                                                         const float* __restrict__ x,
                                                         const float* __restrict__ tw,
                                                         const float* __restrict__ tb,
                                                         const float* __restrict__ pw,
                                                         const float* __restrict__ pb,
                                                         float* __restrict__ out) {
    const int g = threadIdx.x >> 6;          // point group 0..3
    const int e = threadIdx.x & 63;          // channel
    const int p = (blockIdx.x << 2) + g;     // flattened point over B*N

    __shared__ int sidx[4][KK];
    if (e < KK) sidx[g][e] = knn[(size_t)p * KK + e];
    __syncthreads();

    const int b = p >> 12;                   // p / N
    const float* Tb = Tm + (size_t)b * NN * EE;

    float m = Tb[(size_t)sidx[g][0] * EE + e];
#pragma unroll 8
    for (int k = 1; k < KK; ++k)
        m = fmaxf(m, Tb[(size_t)sidx[g][k] * EE + e]);

    const float* xp = x + (size_t)p * DD;
    const float basev = pb[e] + tb[e]
                      + xp[0] * (pw[e]          - tw[e])
                      + xp[1] * (pw[EE + e]     - tw[EE + e])
                      + xp[2] * (pw[2 * EE + e] - tw[2 * EE + e]);

    out[(size_t)p * EE + e] = m + basev;
}

extern "C" void kernel_launch(void* const* d_in, const int* in_sizes, int n_in,
                              void* d_out, int out_size, void* d_ws, size_t ws_size,
                              hipStream_t stream) {
    const float* x  = (const float*)d_in[0];
    const float* tw = (const float*)d_in[1];
    const float* tb = (const float*)d_in[2];
    const float* pw = (const float*)d_in[3];
    const float* pb = (const float*)d_in[4];
    float* out = (float*)d_out;

    char* ws = (char*)d_ws;
    int*   knn = (int*)ws;                                          // B*N*K ints  = 4 MB
    float* Tm  = (float*)(ws + (size_t)BB * NN * KK * sizeof(int)); // B*N*E f32   = 8 MB

    knn_topk_kernel<<<BB * (NN / 16), 512, 0, stream>>>(x, knn);
    proj_kernel<<<(BB * NN) / 64, 256, 0, stream>>>(x, tw, Tm);
    gather_max_kernel<<<(BB * NN) / 4, 256, 0, stream>>>(knn, Tm, x, tw, tb, pw, pb, out);
}